// OptLinker_35296041238831
// MI455X (gfx1250) — compile-verified
//
#include <hip/hip_runtime.h>

// MI455X / gfx1250, wave32. One workgroup (8 waves) per graph.
// Stage 1 (waves 0-2): zc = z@Wc^T + bc, zy = z@Wy^T + by via V_WMMA_F32_16X16X4_F32,
//                      store Ec=exp(-zc), Ey=exp(-zy) to LDS (Ey=1e38 for masked nodes).
// Stage 2 (all waves): out[i,h] = m_i * (sum_j rcp(1 + Ec[i,h]*Ey[j,h])) / denom.

typedef __attribute__((ext_vector_type(2))) float v2f;
typedef __attribute__((ext_vector_type(8))) float v8f;

#define NB   512
#define NV   48
#define HD   32
#define LOG2E 1.4426950408889634f

__global__ __launch_bounds__(256)
void attn_fused_kernel(const float* __restrict__ z,
                       const float* __restrict__ Wc, const float* __restrict__ bc,
                       const float* __restrict__ Wy, const float* __restrict__ by,
                       const int*   __restrict__ mask,
                       float* __restrict__ out)
{
    __shared__ float sEc[NV * HD];   // exp(-zc)
    __shared__ float sEy[NV * HD];   // exp(-zy), 1e38 for invalid nodes
    __shared__ float sMask[NV];

    const int b    = blockIdx.x;
    const int tid  = threadIdx.x;
    const int wave = tid >> 5;
    const int lane = tid & 31;

    if (tid < NV) sMask[tid] = (float)mask[b * NV + tid];
    __syncthreads();

    // ---------------- Stage 1: WMMA GEMMs (waves 0..2, one 16-row tile each) -------------
    if (wave < 3) {
        const int mloc = lane & 15;   // M within tile / N within tile (C/D col)
        const int hi   = lane >> 4;   // half-wave selector

        // A fragments: A[m, k] with k = 4*k0 + 2*hi + {0,1}  (ISA 32-bit A 16x4 layout)
        v2f afrag[8];
        const float* zrow = z + (size_t)(b * NV + wave * 16 + mloc) * HD + hi * 2;
        #pragma unroll
        for (int k0 = 0; k0 < 8; ++k0)
            afrag[k0] = *(const v2f*)(zrow + k0 * 4);

        // Preload ALL weight (B) fragments into distinct registers so every
        // global load issues up-front in clauses and the 4 independent WMMA
        // chains run load-free (no per-WMMA s_wait_loadcnt stalls).
        // B[k,n] = W[n,k] since zc = z @ W^T; lane supplies col n = nt*16+mloc,
        // rows k = 4*k0 + 2*hi + {0,1} -> contiguous float2 in row n of W.
        v2f bfrag[2][2][8];  // [mat][nt][k0]
        #pragma unroll
        for (int nt = 0; nt < 2; ++nt) {
            const float* wcrow = Wc + (nt * 16 + mloc) * HD + hi * 2;
            const float* wyrow = Wy + (nt * 16 + mloc) * HD + hi * 2;
            #pragma unroll
            for (int k0 = 0; k0 < 8; ++k0) {
                bfrag[0][nt][k0] = *(const v2f*)(wcrow + k0 * 4);
                bfrag[1][nt][k0] = *(const v2f*)(wyrow + k0 * 4);
            }
        }

        #pragma unroll
        for (int mat = 0; mat < 2; ++mat) {
            const float* bias = mat ? by : bc;
            float*       dst  = mat ? sEy : sEc;
            #pragma unroll
            for (int nt = 0; nt < 2; ++nt) {
                const int n = nt * 16 + mloc;              // output column
                v8f acc = {0.f, 0.f, 0.f, 0.f, 0.f, 0.f, 0.f, 0.f};
                #pragma unroll
                for (int k0 = 0; k0 < 8; ++k0) {
                    acc = __builtin_amdgcn_wmma_f32_16x16x4_f32(
                        /*neg_a=*/false, afrag[k0],
                        /*neg_b=*/false, bfrag[mat][nt][k0],
                        /*c_mod=*/(short)0, acc,
                        /*reuse_a=*/false, /*reuse_b=*/false);
                }
                const float bv = bias[n];
                #pragma unroll
                for (int v = 0; v < 8; ++v) {
                    const int i   = wave * 16 + hi * 8 + v;          // C/D row
                    float     val = acc[v] + bv;
                    float     e   = __builtin_amdgcn_exp2f(-val * LOG2E);  // exp(-val)
                    if (mat == 1 && sMask[i] == 0.0f) e = 1.0e38f;   // invalid j -> ~0
                    dst[i * HD + n] = e;
                }
            }
        }
    }
    __syncthreads();

    // ---------------- Stage 2: masked pairwise sigmoid reduction -------------------------
    float denom = 0.f;
    #pragma unroll
    for (int j = 0; j < NV; ++j) denom += sMask[j];
    const float invD = __builtin_amdgcn_rcpf(denom);

    const int h  = lane;        // column -> contiguous LDS/global accesses per wave
    const int i0 = wave;        // 0..7
    #pragma unroll
    for (int p = 0; p < 6; ++p) {
        const int   i  = i0 + p * 8;          // 0..47
        const float ec = sEc[i * HD + h];
        float acc = 0.f;
        #pragma unroll 8
        for (int j = 0; j < NV; ++j) {
            // sigmoid(zc+zy) = 1/(1 + Ec*Ey); invalid j: Ey=1e38 -> rcp(huge) ~ 0
            acc += __builtin_amdgcn_rcpf(fmaf(ec, sEy[j * HD + h], 1.0f));
        }
        out[(size_t)(b * NV + i) * HD + h] = sMask[i] * acc * invD;
    }
}

extern "C" void kernel_launch(void* const* d_in, const int* in_sizes, int n_in,
                              void* d_out, int out_size, void* d_ws, size_t ws_size,
                              hipStream_t stream) {
    (void)in_sizes; (void)n_in; (void)out_size; (void)d_ws; (void)ws_size;
    const float* z    = (const float*)d_in[0];
    const float* Wc   = (const float*)d_in[1];
    const float* bc   = (const float*)d_in[2];
    const float* Wy   = (const float*)d_in[3];
    const float* by   = (const float*)d_in[4];
    const int*   mask = (const int*)  d_in[5];
    float* out = (float*)d_out;
    attn_fused_kernel<<<NB, 256, 0, stream>>>(z, Wc, bc, Wy, by, mask, out);
}